// LSTMModel_76270029243095
// MI455X (gfx1250) — compile-verified
//
#include <hip/hip_runtime.h>

typedef __attribute__((ext_vector_type(2))) float v2f;
typedef __attribute__((ext_vector_type(8))) float v8f;

#define WMMA_F32(A, B, C) \
    __builtin_amdgcn_wmma_f32_16x16x4_f32(false, (A), false, (B), (short)0, (C), false, false)

static __device__ __forceinline__ v8f splat8(float s) {
    v8f r;
#pragma unroll
    for (int i = 0; i < 8; ++i) r[i] = s;
    return r;
}

// fast sigmoid: rcp(1 + e^{-x}) -- single v_rcp_f32
static __device__ __forceinline__ float sigm(float x) {
    return __builtin_amdgcn_rcpf(1.0f + __expf(-x));
}
// fast tanh: 1 - 2*rcp(e^{2x}+1); saturates correctly at +/-inf
static __device__ __forceinline__ float tanh_f(float x) {
    float e = __expf(2.0f * x);
    return 1.0f - 2.0f * __builtin_amdgcn_rcpf(e + 1.0f);
}

// One wave per 16 batch rows. Block = 128 threads = 4 waves = 64 rows.
__global__ __launch_bounds__(128)
void lstm_wmma_kernel(const float* __restrict__ x,
                      const float* __restrict__ Wih1, const float* __restrict__ Whh1,
                      const float* __restrict__ bih1, const float* __restrict__ bhh1,
                      const float* __restrict__ Wih2, const float* __restrict__ Whh2,
                      const float* __restrict__ bih2, const float* __restrict__ bhh2,
                      const float* __restrict__ W1,   const float* __restrict__ b1,
                      const float* __restrict__ W2,   const float* __restrict__ b2,
                      float* __restrict__ out, int T)
{
    // per-wave h scratch: 16 rows x 29 cols (odd stride -> conflict-free A reads)
    // cols 0..9  : h1relu ; cols 10..24 : h2 ; cols 25..28 : always zero (K padding)
    __shared__ float HS[4][16 * 29];

    const int lane  = threadIdx.x & 31;
    const int wave  = threadIdx.x >> 5;
    const int j     = lane & 15;              // N index (gate unit) / A row (batch-in-tile)
    const int khalf = (lane < 16) ? 0 : 2;    // A/B K sub-offset per lane half
    const int voff  = (lane < 16) ? 0 : 8;    // C/D row offset per lane half
    const int base  = (blockIdx.x * 4 + wave) * 16;
    const int jrow  = j * 29;
    float* hs = HS[wave];

    // zero the scratch tile (h1relu(-1) = h2(-1) = 0 initial state + K padding)
    for (int i = lane; i < 16 * 29; i += 32) hs[i] = 0.f;

    // ---- per-lane bias splats, built ONCE, used as WMMA C-operand seeds
    v8f bias1s[4], bias2s[4];
#pragma unroll
    for (int gt = 0; gt < 4; ++gt) {
        bias1s[gt] = splat8((j < 10) ? (bih1[gt * 10 + j] + bhh1[gt * 10 + j]) : 0.f);
        bias2s[gt] = splat8((j < 15) ? (bih2[gt * 15 + j] + bhh2[gt * 15 + j]) : 0.f);
    }

    // ---- per-lane B operands, built once, live in VGPRs for all T steps
    v2f B1[3][4];   // LSTM1: K = [Whh1 rows (0..9) | Wih1 (10) | 0 (11)]
#pragma unroll
    for (int kc = 0; kc < 3; ++kc)
#pragma unroll
        for (int gt = 0; gt < 4; ++gt)
#pragma unroll
            for (int p = 0; p < 2; ++p) {
                int K = 4 * kc + khalf + p;
                float w = 0.f;
                if (j < 10) {
                    int g = gt * 10 + j;
                    if (K < 10)       w = Whh1[g * 10 + K];
                    else if (K == 10) w = Wih1[g];
                }
                B1[kc][gt][p] = w;
            }
    v2f B2[7][4];   // LSTM2: K = [Wih2 (0..9) | Whh2 (10..24) | 0 (25..27)]
#pragma unroll
    for (int kc = 0; kc < 7; ++kc)
#pragma unroll
        for (int gt = 0; gt < 4; ++gt)
#pragma unroll
            for (int p = 0; p < 2; ++p) {
                int K = 4 * kc + khalf + p;
                float w = 0.f;
                if (j < 15) {
                    int g = gt * 15 + j;
                    if (K < 10)      w = Wih2[g * 10 + K];
                    else if (K < 25) w = Whh2[g * 15 + (K - 10)];
                }
                B2[kc][gt][p] = w;
            }

    const float* xrow = x + (size_t)(base + j) * (size_t)T;
    v8f c1 = splat8(0.f), c2 = splat8(0.f);   // cell states in C-layout registers

    asm volatile("s_wait_dscnt 0" ::: "memory");

    // ===== prologue: LSTM1 step 0 (h(-1)=0, so A = [0 | x0 | 0], no LDS reads) =====
    {
        float x0 = xrow[0];
        v2f az; az[0] = 0.f; az[1] = 0.f;
        v2f ax; ax[0] = (lane < 16) ? 0.f : x0; ax[1] = 0.f;
        v8f i1 = WMMA_F32(az, B1[0][0], bias1s[0]);
        v8f f1 = WMMA_F32(az, B1[0][1], bias1s[1]);
        v8f g1 = WMMA_F32(az, B1[0][2], bias1s[2]);
        v8f o1 = WMMA_F32(az, B1[0][3], bias1s[3]);
        i1 = WMMA_F32(ax, B1[2][0], i1);
        f1 = WMMA_F32(ax, B1[2][1], f1);
        g1 = WMMA_F32(ax, B1[2][2], g1);
        o1 = WMMA_F32(ax, B1[2][3], o1);
#pragma unroll
        for (int v = 0; v < 8; ++v) {
            float ig = sigm(i1[v]);
            float fg = sigm(f1[v]);
            float gg = tanh_f(g1[v]);
            float og = sigm(o1[v]);
            float c  = fmaf(fg, c1[v], ig * gg);
            c1[v] = c;
            hs[(v + voff) * 29 + j] = fmaxf(og * tanh_f(c), 0.f);  // h1relu(0)
        }
        asm volatile("s_wait_dscnt 0" ::: "memory");
    }

    float xv = xrow[(1 < T) ? 1 : 0];         // x(t+1) for the pipelined LSTM1

    // ===== pipelined scan: iteration t computes LSTM2(t) and LSTM1(t+1) =====
    for (int t = 0; t < T; ++t) {
        float xv_next = xrow[(t + 2 < T) ? (t + 2) : (T - 1)];

        // A operands: a2[0..6] = [h1relu(t) | h2(t-1) | 0]; LSTM1 shares a2[0..1],
        // and its chunk 2 differs only in the upper lane half (x vs h2 cols 10,11).
        v2f a2[7];
#pragma unroll
        for (int kc = 0; kc < 7; ++kc)
#pragma unroll
            for (int p = 0; p < 2; ++p)
                a2[kc][p] = hs[jrow + 4 * kc + khalf + p];
        v2f a1x;
        a1x[0] = (lane < 16) ? a2[2][0] : xv;   // K=8,9 : h1relu | K=10,11 : x(t+1), 0
        a1x[1] = (lane < 16) ? a2[2][1] : 0.f;

        // 8 independent WMMA accumulation chains (i2,f2,g2,o2,i1,f1,g1,o1)
        v8f i2 = WMMA_F32(a2[0], B2[0][0], bias2s[0]);
        v8f f2 = WMMA_F32(a2[0], B2[0][1], bias2s[1]);
        v8f g2 = WMMA_F32(a2[0], B2[0][2], bias2s[2]);
        v8f o2 = WMMA_F32(a2[0], B2[0][3], bias2s[3]);
        v8f i1 = WMMA_F32(a2[0], B1[0][0], bias1s[0]);
        v8f f1 = WMMA_F32(a2[0], B1[0][1], bias1s[1]);
        v8f g1 = WMMA_F32(a2[0], B1[0][2], bias1s[2]);
        v8f o1 = WMMA_F32(a2[0], B1[0][3], bias1s[3]);
#pragma unroll
        for (int kc = 1; kc < 7; ++kc) {
            i2 = WMMA_F32(a2[kc], B2[kc][0], i2);
            f2 = WMMA_F32(a2[kc], B2[kc][1], f2);
            g2 = WMMA_F32(a2[kc], B2[kc][2], g2);
            o2 = WMMA_F32(a2[kc], B2[kc][3], o2);
        }
        i1 = WMMA_F32(a2[1], B1[1][0], i1);
        f1 = WMMA_F32(a2[1], B1[1][1], f1);
        g1 = WMMA_F32(a2[1], B1[1][2], g1);
        o1 = WMMA_F32(a2[1], B1[1][3], o1);
        i1 = WMMA_F32(a1x, B1[2][0], i1);
        f1 = WMMA_F32(a1x, B1[2][1], f1);
        g1 = WMMA_F32(a1x, B1[2][2], g1);
        o1 = WMMA_F32(a1x, B1[2][3], o1);

        // act2(t) -> h2(t) (cols 10..24) ; act1(t+1) -> h1relu(t+1) (cols 0..9)
#pragma unroll
        for (int v = 0; v < 8; ++v) {
            float ig2 = sigm(i2[v]);
            float fg2 = sigm(f2[v]);
            float gg2 = tanh_f(g2[v]);
            float og2 = sigm(o2[v]);
            float cc2 = fmaf(fg2, c2[v], ig2 * gg2);
            c2[v] = cc2;
            hs[(v + voff) * 29 + 10 + j] = og2 * tanh_f(cc2);

            float ig1 = sigm(i1[v]);
            float fg1 = sigm(f1[v]);
            float gg1 = tanh_f(g1[v]);
            float og1 = sigm(o1[v]);
            float cc1 = fmaf(fg1, c1[v], ig1 * gg1);
            c1[v] = cc1;
            hs[(v + voff) * 29 + j] = fmaxf(og1 * tanh_f(cc1), 0.f);
        }
        asm volatile("s_wait_dscnt 0" ::: "memory");

        xv = xv_next;
    }

    // ===== MLP head on h2(T-1): relu(h2 @ W1.T + b1) @ W2.T + b2 =====
    if (lane < 16) {
        float hv[15];
#pragma unroll
        for (int k = 0; k < 15; ++k) hv[k] = hs[jrow + 10 + k];
        float o = b2[0];
#pragma unroll
        for (int u = 0; u < 10; ++u) {
            float z = b1[u];
#pragma unroll
            for (int k = 0; k < 15; ++k) z = fmaf(hv[k], W1[u * 15 + k], z);
            o = fmaf(fmaxf(z, 0.f), W2[u], o);
        }
        out[base + j] = o;
    }
}

extern "C" void kernel_launch(void* const* d_in, const int* in_sizes, int n_in,
                              void* d_out, int out_size, void* d_ws, size_t ws_size,
                              hipStream_t stream) {
    const float* x    = (const float*)d_in[0];
    const float* Wih1 = (const float*)d_in[1];
    const float* Whh1 = (const float*)d_in[2];
    const float* bih1 = (const float*)d_in[3];
    const float* bhh1 = (const float*)d_in[4];
    const float* Wih2 = (const float*)d_in[5];
    const float* Whh2 = (const float*)d_in[6];
    const float* bih2 = (const float*)d_in[7];
    const float* bhh2 = (const float*)d_in[8];
    const float* W1   = (const float*)d_in[9];
    const float* b1   = (const float*)d_in[10];
    const float* W2   = (const float*)d_in[11];
    const float* b2   = (const float*)d_in[12];

    const int B = out_size;              // 4096
    const int T = in_sizes[0] / B;       // 512
    const int blocks = B / 64;           // 4 waves/block * 16 rows/wave

    lstm_wmma_kernel<<<blocks, 128, 0, stream>>>(
        x, Wih1, Whh1, bih1, bhh1, Wih2, Whh2, bih2, bhh2,
        W1, b1, W2, b2, (float*)d_out, T);
}